// SessionGraph_52922587021342
// MI455X (gfx1250) — compile-verified
//
#include <hip/hip_runtime.h>
#include <hip/hip_bf16.h>

// ---------------------------------------------------------------------------
// SR-GNN (SessionGraph) for MI455X / gfx1250.
// B=16, N=50, H=128, V=40000, STEP=2.
// Dominant work: S[(b,n),v] = qt . emb  (8.2 GFLOP f32)  -> V_WMMA_F32_16X16X4_F32
// ---------------------------------------------------------------------------

#define BB 16
#define NN 50
#define HH 128
#define VV 40000
#define PADS 132   // padded LDS row stride (floats) to avoid bank conflicts

typedef __attribute__((ext_vector_type(2))) float v2f;
typedef __attribute__((ext_vector_type(8))) float v8f;

__device__ __forceinline__ float sigmoidf_(float x) { return 1.0f / (1.0f + expf(-x)); }

// ---------------------------------------------------------------------------
// hidden[b,n,:] = emb[items[b,n], :]
__global__ void gather_kernel(const int* __restrict__ items,
                              const float* __restrict__ emb,
                              float* __restrict__ hidden) {
    int bn = blockIdx.x;
    int h  = threadIdx.x;
    hidden[bn * HH + h] = emb[items[bn] * HH + h];
}

// ---------------------------------------------------------------------------
// Generic small linear: Y[m, t] = sum_k X[m,k] * W[t,k] + bias[t]
// grid = M, block = O, dyn-smem = K floats.
__global__ void linear_kernel(const float* __restrict__ X,
                              const float* __restrict__ W,
                              const float* __restrict__ bias,
                              float* __restrict__ Y, int K, int O) {
    extern __shared__ float sx[];
    int m = blockIdx.x, t = threadIdx.x;
    for (int k = t; k < K; k += O) sx[k] = X[m * K + k];
    __syncthreads();
    float acc = bias ? bias[t] : 0.0f;
    const float* wr = &W[t * K];
    for (int k = 0; k < K; ++k) acc += sx[k] * wr[k];
    Y[m * O + t] = acc;
}

// ---------------------------------------------------------------------------
// inputs[b,n, 0:H]   = sum_m A[b,n,m]   * ein [b,m,:] + b_iah
// inputs[b,n, H:2H]  = sum_m A[b,n,N+m] * eout[b,m,:] + b_oah
__global__ void agg_kernel(const float* __restrict__ A,
                           const float* __restrict__ ein,
                           const float* __restrict__ eout,
                           const float* __restrict__ b_iah,
                           const float* __restrict__ b_oah,
                           float* __restrict__ inputs) {
    __shared__ float arow[2 * NN];
    int bn = blockIdx.x;           // b*N + n
    int b  = bn / NN;
    int h  = threadIdx.x;
    for (int m = threadIdx.x; m < 2 * NN; m += blockDim.x)
        arow[m] = A[bn * (2 * NN) + m];
    __syncthreads();
    float s_in = b_iah[h], s_out = b_oah[h];
    for (int m = 0; m < NN; ++m) {
        s_in  += arow[m]      * ein [(b * NN + m) * HH + h];
        s_out += arow[NN + m] * eout[(b * NN + m) * HH + h];
    }
    inputs[bn * (2 * HH) + h]      = s_in;
    inputs[bn * (2 * HH) + HH + h] = s_out;
}

// ---------------------------------------------------------------------------
// GRU gate update (in place on hidden)
__global__ void gru_kernel(const float* __restrict__ gi,
                           const float* __restrict__ gh,
                           float* __restrict__ hidden) {
    int bn = blockIdx.x, h = threadIdx.x;
    float i_r = gi[bn * 3 * HH + h],          h_r = gh[bn * 3 * HH + h];
    float i_i = gi[bn * 3 * HH + HH + h],     h_i = gh[bn * 3 * HH + HH + h];
    float i_n = gi[bn * 3 * HH + 2 * HH + h], h_n = gh[bn * 3 * HH + 2 * HH + h];
    float r = sigmoidf_(i_r + h_r);
    float z = sigmoidf_(i_i + h_i);
    float ng = tanhf(i_n + r * h_n);
    float hp = hidden[bn * HH + h];
    hidden[bn * HH + h] = ng + z * (hp - ng);
}

// ---------------------------------------------------------------------------
// ht[b,:] = hidden[b, len(b)-1, :]
__global__ void ht_kernel(const int* __restrict__ mask,
                          const float* __restrict__ hidden,
                          float* __restrict__ ht) {
    int b = blockIdx.x, h = threadIdx.x;
    int len = 0;
    for (int n = 0; n < NN; ++n) len += mask[b * NN + n];
    ht[b * HH + h] = hidden[(b * NN + len - 1) * HH + h];
}

// ---------------------------------------------------------------------------
// logits[b,n] = sum_h sigmoid(q1[b,h] + q2[b,n,h]) * w3[h]
__global__ void alpha_logits_kernel(const float* __restrict__ q1,
                                    const float* __restrict__ q2,
                                    const float* __restrict__ w3,
                                    float* __restrict__ logits) {
    __shared__ float red[HH];
    int bn = blockIdx.x;
    int b  = bn / NN;
    int h  = threadIdx.x;
    float x = q1[b * HH + h] + q2[bn * HH + h];
    red[h] = sigmoidf_(x) * w3[h];
    __syncthreads();
    for (int off = HH / 2; off > 0; off >>= 1) {
        if (h < off) red[h] += red[h + off];
        __syncthreads();
    }
    if (h == 0) logits[bn] = red[0];
}

// ---------------------------------------------------------------------------
// a_cat[b, 0:H] = sum_n softmax_n(logits | mask) * hidden[b,n,:]
// a_cat[b, H:2H] = ht[b,:]
__global__ void alpha_apply_kernel(const float* __restrict__ logits,
                                   const int* __restrict__ mask,
                                   const float* __restrict__ hidden,
                                   const float* __restrict__ ht,
                                   float* __restrict__ a_cat) {
    int b = blockIdx.x, h = threadIdx.x;
    float mx = -3.0e38f;
    for (int n = 0; n < NN; ++n)
        if (mask[b * NN + n]) mx = fmaxf(mx, logits[b * NN + n]);
    float Z = 0.0f;
    for (int n = 0; n < NN; ++n)
        if (mask[b * NN + n]) Z += expf(logits[b * NN + n] - mx);
    float acc = 0.0f;
    for (int n = 0; n < NN; ++n)
        if (mask[b * NN + n])
            acc += (expf(logits[b * NN + n] - mx) / Z) * hidden[(b * NN + n) * HH + h];
    a_cat[b * (2 * HH) + h]      = acc;
    a_cat[b * (2 * HH) + HH + h] = ht[b * HH + h];
}

// ---------------------------------------------------------------------------
// hm[b,n,:] = hidden[b,n,:] * mask[b,n]
__global__ void maskmul_kernel(const float* __restrict__ hidden,
                               const int* __restrict__ mask,
                               float* __restrict__ hm) {
    int bn = blockIdx.x, h = threadIdx.x;
    hm[bn * HH + h] = hidden[bn * HH + h] * (float)mask[bn];
}

// ---------------------------------------------------------------------------
// Fused scoring kernel (the 8.2 GFLOP part), f32 WMMA on the matrix pipe.
//   s[n, v]      = qt[b,n] . emb[v]          (v = 1 + global col)
//   p            = softmax over valid n of s
//   scores[b,v]  = sum_n p_n * s_n + a[b] . emb[v]
//
// One workgroup = (batch b, 64-column slab of v). QT (25.6 KB), mask and avec
// are staged to LDS ONCE per block; the block then sweeps 4 column-tiles of 16,
// re-staging only the 8 KB emb tile each time (4x less QT traffic than a
// tile-per-block layout). 4 waves, each owning a 16-row M-tile; K=128 swept in
// 32 x V_WMMA_F32_16X16X4_F32 (full fp32 precision on the matrix pipe).
__global__ __launch_bounds__(128)
void scores_kernel(const float* __restrict__ qt,    // (B,N,H)
                   const float* __restrict__ emb,   // (V,H)
                   const int*   __restrict__ mask,  // (B,N)
                   const float* __restrict__ avec,  // (B,H)
                   float* __restrict__ out) {       // (B, V-1)
    __shared__ float sQT[64 * PADS];   // padded-stride QT rows for this batch
    __shared__ float sE [16 * PADS];   // emb tile (16 candidate items)
    __shared__ float sS [64 * 16];     // S tile (rows n x cols v)
    __shared__ float sD [16 * 9];      // per-column partial a.e dots (pad 9)
    __shared__ float sAvec[HH];
    __shared__ int   sMask[NN];

    const int b   = blockIdx.y;
    const int tid = threadIdx.x;

    // Stage QT once (zero-pad rows >= N), 128-bit loads.
    for (int i = tid; i < 64 * HH / 4; i += 128) {
        int r = (i * 4) >> 7, h = (i * 4) & 127;
        float4 val = make_float4(0.f, 0.f, 0.f, 0.f);
        if (r < NN) val = *(const float4*)&qt[(b * NN + r) * HH + h];
        *(float4*)&sQT[r * PADS + h] = val;
    }
    if (tid < HH) sAvec[tid] = avec[b * HH + tid];
    if (tid < NN) sMask[tid] = mask[b * NN + tid];

    const int wave = tid >> 5;          // 0..3 -> M-tile (16 rows each)
    const int lane = tid & 31;
    const int rc   = lane & 15;         // A: row, B: column
    const int hi   = lane >> 4;         // K-half selector (K pairs {0,1} / {2,3})
    const float* aRow = &sQT[(wave * 16 + rc) * PADS];
    const float* bRow = &sE [rc * PADS];

    for (int ct = 0; ct < 4; ++ct) {
        const int v0 = blockIdx.x * 64 + ct * 16;   // base column in V-1 space
        __syncthreads();   // previous-iter readers done (also covers QT staging)

        // Stage this column-tile's emb rows, 128-bit loads.
        for (int i = tid; i < 16 * HH / 4; i += 128) {
            int r = (i * 4) >> 7, h = (i * 4) & 127;
            int v = v0 + r + 1; if (v > VV - 1) v = VV - 1;   // clamp; store guarded
            *(float4*)&sE[r * PADS + h] = *(const float4*)&emb[(size_t)v * HH + h];
        }
        // Pull the next tile toward the caches while this one computes.
        int vnext = v0 + 17; if (vnext > VV - 1) vnext = VV - 1;
        __builtin_prefetch(&emb[(size_t)vnext * HH], 0, 1);
        __syncthreads();

        v8f c = {};
        #pragma unroll 8
        for (int k0 = 0; k0 < HH; k0 += 4) {
            int kk = k0 + (hi << 1);
            v2f af; af.x = aRow[kk]; af.y = aRow[kk + 1];
            v2f bf; bf.x = bRow[kk]; bf.y = bRow[kk + 1];
            c = __builtin_amdgcn_wmma_f32_16x16x4_f32(false, af, false, bf,
                                                      (short)0, c, false, false);
        }
        #pragma unroll
        for (int g = 0; g < 8; ++g)
            sS[(wave * 16 + g + hi * 8) * 16 + rc] = c[g];

        // All 128 threads: partial dots for a . emb[v] (8 x 16-dim chunks/col).
        {
            int col = tid & 15, part = tid >> 4;
            float pd = 0.0f;
            const float* eRow = &sE[col * PADS + part * 16];
            const float* aV   = &sAvec[part * 16];
            #pragma unroll
            for (int h = 0; h < 16; ++h) pd += aV[h] * eRow[h];
            sD[col * 9 + part] = pd;
        }
        __syncthreads();

        // Epilogue: masked softmax over n + score, one thread per column.
        if (tid < 16) {
            int vglob = v0 + tid;                    // index into V-1
            if (vglob < VV - 1) {
                float mx = -3.0e38f;
                for (int n = 0; n < NN; ++n)
                    if (sMask[n]) mx = fmaxf(mx, sS[n * 16 + tid]);
                float Z = 0.0f, SW = 0.0f;
                for (int n = 0; n < NN; ++n)
                    if (sMask[n]) {
                        float s = sS[n * 16 + tid];
                        float e = expf(s - mx);
                        Z += e; SW += e * s;
                    }
                float ad = 0.0f;
                #pragma unroll
                for (int p = 0; p < 8; ++p) ad += sD[tid * 9 + p];
                out[(size_t)b * (VV - 1) + vglob] = SW / Z + ad;
            }
        }
    }
}

// ---------------------------------------------------------------------------
extern "C" void kernel_launch(void* const* d_in, const int* in_sizes, int n_in,
                              void* d_out, int out_size, void* d_ws, size_t ws_size,
                              hipStream_t stream) {
    const int*   items  = (const int*)d_in[0];
    const float* A      = (const float*)d_in[1];
    const int*   mask   = (const int*)d_in[2];
    const float* emb    = (const float*)d_in[3];
    const float* w_ih   = (const float*)d_in[4];
    const float* w_hh   = (const float*)d_in[5];
    const float* b_ih   = (const float*)d_in[6];
    const float* b_hh   = (const float*)d_in[7];
    const float* b_iah  = (const float*)d_in[8];
    const float* b_oah  = (const float*)d_in[9];
    const float* W_ein  = (const float*)d_in[10];
    const float* b_ein  = (const float*)d_in[11];
    const float* W_eout = (const float*)d_in[12];
    const float* b_eout = (const float*)d_in[13];
    const float* W_one  = (const float*)d_in[14];
    const float* b_one  = (const float*)d_in[15];
    const float* W_two  = (const float*)d_in[16];
    const float* b_two  = (const float*)d_in[17];
    const float* w_three= (const float*)d_in[18];
    const float* W_tr   = (const float*)d_in[19];
    const float* b_tr   = (const float*)d_in[20];
    const float* W_t    = (const float*)d_in[21];

    // Workspace carve-up (floats); ~5.8 MB total.
    float* ws     = (float*)d_ws;
    float* hidden = ws;                      // B*N*H      = 102400
    float* ein    = hidden + BB * NN * HH;   // 102400
    float* eout   = ein    + BB * NN * HH;   // 102400
    float* inputs = eout   + BB * NN * HH;   // B*N*2H     = 204800
    float* gi     = inputs + BB * NN * 2 * HH;   // B*N*3H = 307200
    float* gh     = gi     + BB * NN * 3 * HH;   // 307200
    float* q1     = gh     + BB * NN * 3 * HH;   // B*H   = 2048
    float* q2     = q1     + BB * HH;            // 102400
    float* logits = q2     + BB * NN * HH;       // 800
    float* htb    = logits + BB * NN;            // 2048
    float* a_cat  = htb    + BB * HH;            // 4096
    float* a_vec  = a_cat  + BB * 2 * HH;        // 2048
    float* hm     = a_vec  + BB * HH;            // 102400
    float* qt     = hm     + BB * NN * HH;       // 102400

    const int M = BB * NN;  // 800

    gather_kernel<<<M, HH, 0, stream>>>(items, emb, hidden);

    for (int step = 0; step < 2; ++step) {
        linear_kernel<<<M, HH, HH * sizeof(float), stream>>>(hidden, W_ein, b_ein, ein, HH, HH);
        linear_kernel<<<M, HH, HH * sizeof(float), stream>>>(hidden, W_eout, b_eout, eout, HH, HH);
        agg_kernel<<<M, HH, 0, stream>>>(A, ein, eout, b_iah, b_oah, inputs);
        linear_kernel<<<M, 3 * HH, 2 * HH * sizeof(float), stream>>>(inputs, w_ih, b_ih, gi, 2 * HH, 3 * HH);
        linear_kernel<<<M, 3 * HH, HH * sizeof(float), stream>>>(hidden, w_hh, b_hh, gh, HH, 3 * HH);
        gru_kernel<<<M, HH, 0, stream>>>(gi, gh, hidden);
    }

    ht_kernel<<<BB, HH, 0, stream>>>(mask, hidden, htb);
    linear_kernel<<<BB, HH, HH * sizeof(float), stream>>>(htb, W_one, b_one, q1, HH, HH);
    linear_kernel<<<M, HH, HH * sizeof(float), stream>>>(hidden, W_two, b_two, q2, HH, HH);
    alpha_logits_kernel<<<M, HH, 0, stream>>>(q1, q2, w_three, logits);
    alpha_apply_kernel<<<BB, HH, 0, stream>>>(logits, mask, hidden, htb, a_cat);
    linear_kernel<<<BB, HH, 2 * HH * sizeof(float), stream>>>(a_cat, W_tr, b_tr, a_vec, 2 * HH, HH);
    maskmul_kernel<<<M, HH, 0, stream>>>(hidden, mask, hm);
    linear_kernel<<<M, HH, HH * sizeof(float), stream>>>(hm, W_t, nullptr, qt, HH, HH);

    // 64 columns per block: grid = (ceil((V-1)/64), B)
    dim3 sgrid((VV - 1 + 63) / 64, BB);
    scores_kernel<<<sgrid, 128, 0, stream>>>(qt, emb, mask, a_vec, (float*)d_out);
}